// KWinnersTakeAll_54176717471901
// MI455X (gfx1250) — compile-verified
//
#include <hip/hip_runtime.h>

#define D       16384
#define K_RANK  820u        // ceil(0.05 * 16384)
#define BLOCK   512
#define WAVES   (BLOCK / 32)
#define EPT     (D / BLOCK)  // 32 elements per thread
#define VPT     (EPT / 4)    // 8 float4 per thread

#if defined(__has_builtin)
#if __has_builtin(__builtin_amdgcn_global_load_async_to_lds_b128)
#define HAS_ASYNC_LDS 1
#endif
#endif

typedef int v4i __attribute__((ext_vector_type(4)));
typedef __attribute__((address_space(1))) v4i GV4;   // global int4
typedef __attribute__((address_space(3))) v4i LV4;   // LDS int4

__device__ __forceinline__ unsigned f2key(unsigned u) {
  // monotonic float -> uint mapping (ascending order preserved)
  return u ^ ((unsigned)((int)u >> 31) | 0x80000000u);
}

__device__ __forceinline__ float key2f(unsigned k) {
  unsigned u = (k & 0x80000000u) ? (k ^ 0x80000000u) : ~k;
  return __uint_as_float(u);
}

__device__ __forceinline__ void wait_async_zero() {
#if defined(__has_builtin)
#if __has_builtin(__builtin_amdgcn_s_wait_asynccnt)
  __builtin_amdgcn_s_wait_asynccnt(0);
#else
  asm volatile("s_wait_asynccnt 0x0" ::: "memory");
#endif
#else
  asm volatile("s_wait_asynccnt 0x0" ::: "memory");
#endif
}

__global__ __launch_bounds__(BLOCK) void kwta_kernel(const float* __restrict__ x,
                                                     float* __restrict__ out) {
  __shared__ float    s_vals[D];               // 64 KB row staging
  __shared__ unsigned s_hist[256 * WAVES];     // 16 KB per-wave histograms (bank-spread)
  __shared__ unsigned s_cnt[256];
  __shared__ unsigned s_prefix, s_k, s_cntd, s_max;

  const int tid  = threadIdx.x;
  const int wv   = tid >> 5;
  const int lane = tid & 31;
  const size_t row = blockIdx.x;

  const float4* __restrict__ src4 = (const float4*)(x + row * (size_t)D);
  float4* __restrict__ dst4 = (float4*)(out + row * (size_t)D);
  float4* s_vals4 = (float4*)s_vals;

  // ---- Stage row into LDS (async global->LDS path on gfx1250) ----
#if defined(HAS_ASYNC_LDS)
  {
    GV4* g = (GV4*)(x + row * (size_t)D);
    LV4* l = (LV4*)s_vals;
#pragma unroll
    for (int j = 0; j < VPT; ++j) {
      const int idx = j * BLOCK + tid;
      __builtin_amdgcn_global_load_async_to_lds_b128(g + idx, l + idx, 0, 0);
    }
  }
#else
#pragma unroll
  for (int j = 0; j < VPT; ++j) {
    const int idx = j * BLOCK + tid;
    s_vals4[idx] = src4[idx];
  }
#endif

  if (tid == 0) { s_prefix = 0u; s_k = K_RANK; }
#if defined(HAS_ASYNC_LDS)
  wait_async_zero();
#endif
  __syncthreads();

  // ---- 4-pass MSB radix select for the K_RANK-th largest key ----
  for (int pass = 0; pass < 4; ++pass) {
    const int shift = 24 - 8 * pass;
    const unsigned pmask = pass ? (0xFFFFFFFFu << (shift + 8)) : 0u;

    // clear per-wave histograms
#pragma unroll
    for (int j = 0; j < (256 * WAVES) / BLOCK; ++j)
      s_hist[j * BLOCK + tid] = 0u;
    __syncthreads();

    const unsigned pref = s_prefix;
#pragma unroll
    for (int j = 0; j < VPT; ++j) {
      const int idx = j * BLOCK + tid;
      const float4 v = s_vals4[idx];
      unsigned kk;
      kk = f2key(__float_as_uint(v.x));
      if ((kk & pmask) == pref) atomicAdd(&s_hist[((kk >> shift) & 0xFFu) * WAVES + wv], 1u);
      kk = f2key(__float_as_uint(v.y));
      if ((kk & pmask) == pref) atomicAdd(&s_hist[((kk >> shift) & 0xFFu) * WAVES + wv], 1u);
      kk = f2key(__float_as_uint(v.z));
      if ((kk & pmask) == pref) atomicAdd(&s_hist[((kk >> shift) & 0xFFu) * WAVES + wv], 1u);
      kk = f2key(__float_as_uint(v.w));
      if ((kk & pmask) == pref) atomicAdd(&s_hist[((kk >> shift) & 0xFFu) * WAVES + wv], 1u);
    }
    __syncthreads();

    // reduce per-wave histograms -> s_cnt[256]
    if (tid < 256) {
      unsigned t = 0;
#pragma unroll
      for (int w = 0; w < WAVES; ++w) t += s_hist[tid * WAVES + w];
      s_cnt[tid] = t;
    }
    __syncthreads();

    // one wave: shuffle-based suffix scan over 256 bins, pick target digit
    if (tid < 32) {
      unsigned binv[8];
      unsigned p = 0;
#pragma unroll
      for (int i = 0; i < 8; ++i) { binv[i] = s_cnt[lane * 8 + i]; p += binv[i]; }
      unsigned s = p;
#pragma unroll
      for (int off = 1; off < 32; off <<= 1) {
        unsigned t = __shfl_down(s, off, 32);
        if (lane + off < 32) s += t;
      }
      unsigned snext = __shfl_down(s, 1, 32);
      if (lane == 31) snext = 0u;
      const unsigned kcur = s_k;
      if (s >= kcur && snext < kcur) {
        unsigned running = snext;
#pragma unroll
        for (int i = 7; i >= 0; --i) {
          const unsigned c = binv[i];
          if (running + c >= kcur) {
            s_prefix = pref | ((unsigned)(lane * 8 + i) << shift);
            s_k = kcur - running;   // rank within the equal-digit partition
            s_cntd = c;             // partition size (== #equal keys on last pass)
            break;
          }
          running += c;
        }
      }
    }
    __syncthreads();
  }

  const unsigned key820 = s_prefix;  // exact bit pattern of 820th largest
  const unsigned kfin   = s_k;
  const unsigned ceq    = s_cntd;

  // 821st largest: tie with key820, or max key strictly below it
  const bool tie = (ceq > kfin);   // c_gt + c_eq >= 821
  unsigned key821;
  if (tie) {
    key821 = key820;
  } else {
    if (tid == 0) s_max = 0u;
    __syncthreads();
    unsigned m = 0u;
#pragma unroll
    for (int j = 0; j < VPT; ++j) {
      const int idx = j * BLOCK + tid;
      const float4 v = s_vals4[idx];
      unsigned kk;
      kk = f2key(__float_as_uint(v.x)); if (kk < key820 && kk > m) m = kk;
      kk = f2key(__float_as_uint(v.y)); if (kk < key820 && kk > m) m = kk;
      kk = f2key(__float_as_uint(v.z)); if (kk < key820 && kk > m) m = kk;
      kk = f2key(__float_as_uint(v.w)); if (kk < key820 && kk > m) m = kk;
    }
#pragma unroll
    for (int off = 16; off > 0; off >>= 1) {
      const unsigned t = __shfl_xor(m, off, 32);
      m = (t > m) ? t : m;
    }
    if (lane == 0) atomicMax(&s_max, m);
    __syncthreads();
    key821 = s_max;
  }

  const float thr = (key2f(key820) + key2f(key821)) * 0.5f;

  // ---- emit mask (b128 stores) ----
#pragma unroll
  for (int j = 0; j < VPT; ++j) {
    const int idx = j * BLOCK + tid;
    const float4 v = s_vals4[idx];
    float4 o;
    o.x = (v.x > thr) ? 1.0f : 0.0f;
    o.y = (v.y > thr) ? 1.0f : 0.0f;
    o.z = (v.z > thr) ? 1.0f : 0.0f;
    o.w = (v.w > thr) ? 1.0f : 0.0f;
    dst4[idx] = o;
  }
}

extern "C" void kernel_launch(void* const* d_in, const int* in_sizes, int n_in,
                              void* d_out, int out_size, void* d_ws, size_t ws_size,
                              hipStream_t stream) {
  (void)n_in; (void)d_ws; (void)ws_size; (void)out_size;
  const float* x = (const float*)d_in[0];
  float* out = (float*)d_out;
  const int rows = in_sizes[0] / D;   // 4096
  kwta_kernel<<<rows, BLOCK, 0, stream>>>(x, out);
}